// PyTorchPatternMatcher_64252710748373
// MI455X (gfx1250) — compile-verified
//
#include <hip/hip_runtime.h>
#include <hip/hip_bf16.h>
#include <stdint.h>
#include <math.h>

// ---------------------------------------------------------------------------
// Pattern matcher, CDNA5 (gfx1250) version.
//
// max_diff_L = max_i |x[i] - x[i mod L]|.  All 20 candidate L divide
// M = 110880 = lcm(candidates), and M | N with N/M = 512.  So a single
// streaming pass computing per-residue (mod M) max/min suffices; the 20
// candidate reductions then run over just M elements.
//
// Pass 1 uses the CDNA5 async-tensor path: global_load_async_to_lds_b128
// into an 8-deep LDS ring per thread, throttled with s_wait_asynccnt,
// consumed via ds_load_b128.  Memory-bound: 227 MB @ 23.3 TB/s ~ 10 us.
// ---------------------------------------------------------------------------

namespace {
constexpr int       K_CAND = 20;
constexpr int       LENS[K_CAND] = {4,5,6,7,8,9,10,11,12,14,15,16,18,20,21,22,24,28,30,32};
constexpr int       M_RES  = 110880;           // lcm of candidate lengths
constexpr long long N_TOT  = 56770560LL;       // 55440 * 1024
constexpr int       ROWS   = 512;              // N_TOT / M_RES
constexpr int       NCHUNK = 16;               // row-splitting for parallelism
constexpr int       RPC    = ROWS / NCHUNK;    // 32 rows per block
constexpr int       DEPTH  = 8;                // async LDS ring depth
constexpr int       TPB    = 256;
constexpr int       NTHREADS_RES = M_RES / 4;  // 27720; each thread owns 4 residues
} // namespace

// Sign-aware float atomic max/min via integer atomics (valid for any floats
// when the slot is initialized to -inf / +inf).
__device__ __forceinline__ void atomic_max_f32(float* a, float v) {
  if (v >= 0.0f) atomicMax((int*)a, __float_as_int(v));
  else           atomicMin((unsigned int*)a, __float_as_uint(v));
}
__device__ __forceinline__ void atomic_min_f32(float* a, float v) {
  if (v >= 0.0f) atomicMin((int*)a, __float_as_int(v));
  else           atomicMax((unsigned int*)a, __float_as_uint(v));
}

__global__ void __launch_bounds__(TPB)
init_kernel(float* __restrict__ maxv, float* __restrict__ minv,
            float* __restrict__ accum) {
  const int i = blockIdx.x * TPB + threadIdx.x;
  if (i < M_RES) {
    maxv[i] = -INFINITY;
    minv[i] =  INFINITY;
  }
  if (i < K_CAND) accum[i] = 0.0f;   // candidate diffs are always >= 0
}

// Pass 1: per-residue max/min over this block's 32-row slice, streamed
// through LDS with CDNA5 async loads.
__global__ void __launch_bounds__(TPB)
minmax_kernel(const float* __restrict__ x,
              float* __restrict__ maxv, float* __restrict__ minv) {
  __shared__ float4 stage[DEPTH][TPB];   // 32 KB ring, [slot][tid] => no bank conflicts

  const int t = blockIdx.x * TPB + threadIdx.x;
  if (t >= NTHREADS_RES) return;         // tail of last residue block
  const int r0   = t << 2;               // 4 consecutive residues per thread
  const int row0 = blockIdx.y * RPC;

  const unsigned lds_base =
      (unsigned)(uintptr_t)(void*)&stage[0][threadIdx.x];   // AS3 byte offset
  unsigned long long ga =
      (unsigned long long)(uintptr_t)x +
      ((unsigned long long)row0 * (unsigned long long)M_RES +
       (unsigned long long)r0) * 4ull;
  const unsigned long long gstep = (unsigned long long)M_RES * 4ull;

  // Prologue: arm the async pipeline (DEPTH rows in flight per wave).
#pragma unroll
  for (int i = 0; i < DEPTH; ++i) {
    const unsigned lo = lds_base + (unsigned)(i * TPB * 16);
    asm volatile("global_load_async_to_lds_b128 %0, %1, off"
                 :: "v"(lo), "v"(ga) : "memory");
    ga += gstep;
  }

  float4 mx = make_float4(-INFINITY, -INFINITY, -INFINITY, -INFINITY);
  float4 mn = make_float4( INFINITY,  INFINITY,  INFINITY,  INFINITY);

  // Steady state: wait for oldest row, consume it, re-arm its slot.
  for (int i = 0; i < RPC - DEPTH; ++i) {
    asm volatile("s_wait_asynccnt 7" ::: "memory");      // oldest landed in LDS
    const float4 v = stage[i & (DEPTH - 1)][threadIdx.x];// ds_load_b128
    asm volatile("s_wait_dscnt 0" ::: "memory");         // slot read complete
    const unsigned lo = lds_base + (unsigned)((i & (DEPTH - 1)) * TPB * 16);
    asm volatile("global_load_async_to_lds_b128 %0, %1, off"
                 :: "v"(lo), "v"(ga) : "memory");
    ga += gstep;
    mx.x = fmaxf(mx.x, v.x); mx.y = fmaxf(mx.y, v.y);
    mx.z = fmaxf(mx.z, v.z); mx.w = fmaxf(mx.w, v.w);
    mn.x = fminf(mn.x, v.x); mn.y = fminf(mn.y, v.y);
    mn.z = fminf(mn.z, v.z); mn.w = fminf(mn.w, v.w);
  }

  // Drain: everything outstanding has been issued; wait once, consume rest.
  asm volatile("s_wait_asynccnt 0" ::: "memory");
#pragma unroll
  for (int i = RPC - DEPTH; i < RPC; ++i) {
    const float4 v = stage[i & (DEPTH - 1)][threadIdx.x];
    mx.x = fmaxf(mx.x, v.x); mx.y = fmaxf(mx.y, v.y);
    mx.z = fmaxf(mx.z, v.z); mx.w = fmaxf(mx.w, v.w);
    mn.x = fminf(mn.x, v.x); mn.y = fminf(mn.y, v.y);
    mn.z = fminf(mn.z, v.z); mn.w = fminf(mn.w, v.w);
  }

  atomic_max_f32(&maxv[r0 + 0], mx.x); atomic_min_f32(&minv[r0 + 0], mn.x);
  atomic_max_f32(&maxv[r0 + 1], mx.y); atomic_min_f32(&minv[r0 + 1], mn.y);
  atomic_max_f32(&maxv[r0 + 2], mx.z); atomic_min_f32(&minv[r0 + 2], mn.z);
  atomic_max_f32(&maxv[r0 + 3], mx.w); atomic_min_f32(&minv[r0 + 3], mn.w);
}

// Pass 2: fold M residues into the 20 candidate max-diffs.
__global__ void __launch_bounds__(TPB)
diff_kernel(const float* __restrict__ x,
            const float* __restrict__ maxv, const float* __restrict__ minv,
            float* __restrict__ accum) {
  __shared__ float sfirst[32];
  __shared__ int   sred[K_CAND];
  if (threadIdx.x < 32)     sfirst[threadIdx.x] = x[threadIdx.x];
  if (threadIdx.x < K_CAND) sred[threadIdx.x]   = 0;   // bits of +0.0f
  __syncthreads();

  float acc[K_CAND];
#pragma unroll
  for (int k = 0; k < K_CAND; ++k) acc[k] = 0.0f;

  for (int r = blockIdx.x * TPB + threadIdx.x; r < M_RES; r += gridDim.x * TPB) {
    const float hi = maxv[r];
    const float lo = minv[r];
#pragma unroll
    for (int k = 0; k < K_CAND; ++k) {
      const int   L = LENS[k];               // constant => magic-number modulus
      const float p = sfirst[r % L];
      const float c = fmaxf(hi - p, p - lo); // == max_i |x_i - p| over class r
      acc[k] = fmaxf(acc[k], c);
    }
  }
#pragma unroll
  for (int k = 0; k < K_CAND; ++k)
    atomicMax(&sred[k], __float_as_int(acc[k]));       // values >= 0
  __syncthreads();
  if (threadIdx.x < K_CAND)
    atomicMax((int*)&accum[threadIdx.x], sred[threadIdx.x]);
}

// Pass 3: scores, gating, first-occurrence argmax (matches jnp.argmax).
__global__ void finalize_kernel(const float* __restrict__ accum,
                                float* __restrict__ out) {
  if (threadIdx.x != 0 || blockIdx.x != 0) return;
  float eff[K_CAND];
  for (int k = 0; k < K_CAND; ++k) {
    const float md = accum[k];
    out[k] = md;
    const int L = LENS[k];
    const double raw = (double)(N_TOT / (long long)L) / (double)L;
    const float e = (md < 0.01f) ? (float)raw : 0.0f;
    out[K_CAND + k] = e;
    eff[k] = e;
  }
  float best = eff[0];
  int   bidx = 0;
  for (int k = 1; k < K_CAND; ++k)
    if (eff[k] > best) { best = eff[k]; bidx = k; }
  out[2 * K_CAND]     = (float)bidx;
  out[2 * K_CAND + 1] = best;
}

extern "C" void kernel_launch(void* const* d_in, const int* in_sizes, int n_in,
                              void* d_out, int out_size, void* d_ws, size_t ws_size,
                              hipStream_t stream) {
  (void)in_sizes; (void)n_in; (void)out_size; (void)ws_size;
  const float* x   = (const float*)d_in[0];
  float*       out = (float*)d_out;

  float* maxv  = (float*)d_ws;        // [M_RES]
  float* minv  = maxv + M_RES;        // [M_RES]
  float* accum = minv + M_RES;        // [K_CAND]   (~887 KB total)

  init_kernel<<<(M_RES + TPB - 1) / TPB, TPB, 0, stream>>>(maxv, minv, accum);

  dim3 grid((NTHREADS_RES + TPB - 1) / TPB, NCHUNK);   // 109 x 16 blocks
  minmax_kernel<<<grid, TPB, 0, stream>>>(x, maxv, minv);

  diff_kernel<<<96, TPB, 0, stream>>>(x, maxv, minv, accum);

  finalize_kernel<<<1, 32, 0, stream>>>(accum, out);
}